// DualBranchCMA_39178691674140
// MI455X (gfx1250) — compile-verified
//
#include <hip/hip_runtime.h>
#include <math.h>
#include <stdint.h>

// Problem constants (from reference): B=32, C1=256, C2=512, H=W=48
#define NB   32
#define C1N  256
#define C2N  512
#define WDIM 48
#define HWN  2304              // 48*48
#define KCONV (C1N * 9)        // 2304

#define LDS_PITCH 68           // 64 payload floats + 4 pad floats (bank-conflict-free)
#define TILE_F    (16 * LDS_PITCH)

typedef __attribute__((ext_vector_type(2))) float v2f;
typedef __attribute__((ext_vector_type(8))) float v8f;
typedef __attribute__((ext_vector_type(4))) unsigned int v4u;
typedef __attribute__((ext_vector_type(4))) int v4i;
typedef __attribute__((ext_vector_type(8))) int v8i;

// D = A(16x4) * B(4x16) + C  — fp32 WMMA, full-precision path.
__device__ __forceinline__ v8f wmma4(v2f a, v2f b, v8f c) {
  return __builtin_amdgcn_wmma_f32_16x16x4_f32(
      /*neg_a=*/false, a, /*neg_b=*/false, b,
      /*c_mod=*/(short)0, c, /*reuse_a=*/false, /*reuse_b=*/false);
}

__device__ __forceinline__ int wave_id() {
  return __builtin_amdgcn_readfirstlane((int)(threadIdx.x >> 5));
}

// The TDM writes LDS behind the compiler's back (the LDS address is passed as
// a descriptor *value*, so the __shared__ array looks never-written and reads
// from it would fold to undef — seen in round-2 asm). This zero-instruction
// asm escapes the array address and clobbers memory, forcing real ds_loads.
__device__ __forceinline__ void lds_publish(const void* p) {
  asm volatile("" ::"v"((unsigned long long)(uintptr_t)p) : "memory");
}

// ---------------------------------------------------------------------------
// TDM: DMA a 16-row x 64-col fp32 tile from a row-major [th x tw] tensor into
// LDS at byte offset lds_byte_off. pad_enable inserts 4 dwords after every 64
// dwords -> LDS row pitch = 68 floats (16 rows on distinct bank groups).
// Descriptor layout per CDNA5 ISA ch.8 (D# groups 0/1; groups 2/3 zero = 2D).
// ---------------------------------------------------------------------------
__device__ __forceinline__ void tdm_tile_16x64(const float* tile,
                                               unsigned lds_byte_off,
                                               unsigned tw, unsigned th) {
  const unsigned long long ga = (unsigned long long)(uintptr_t)tile;
  v4u g0;
  g0.x = 1u;                                   // count=1 (valid), user mode
  g0.y = lds_byte_off;                         // lds_addr (bytes)
  g0.z = (unsigned)ga;                         // global_addr[31:0]
  g0.w = (unsigned)(ga >> 32) | (2u << 30);    // global_addr[56:32] | type=2
  v8i g1;
  g1[0] = (int)((2u << 16) |                   // data_size = 4 bytes
                (1u << 20) |                   // pad_enable
                (5u << 22) |                   // pad_interval: 64 dwords
                (3u << 25));                   // pad_amount:   4 dwords
  g1[1] = (int)(tw << 16);                     // tensor_dim0[15:0] (cols)
  g1[2] = (int)(th << 16);                     // tensor_dim1[15:0] (rows)
  g1[3] = (int)(64u << 16);                    // tile_dim0 = 64
  g1[4] = 16;                                  // tile_dim1 = 16 (tile_dim2=0)
  g1[5] = (int)tw;                             // tensor_dim0_stride[31:0]
  g1[6] = 0;                                   // stride hi / dim1_stride lo
  g1[7] = 0;
  v4i z4 = {0, 0, 0, 0};
#if __clang_major__ >= 23
  v8i z8 = {0, 0, 0, 0, 0, 0, 0, 0};
  __builtin_amdgcn_tensor_load_to_lds(g0, g1, z4, z4, z8, 0);
#else
  __builtin_amdgcn_tensor_load_to_lds(g0, g1, z4, z4, 0);
#endif
}

// ---------------------------------------------------------------------------
// Kernel 1: semantic 1x1 projection.  sem[b] = proj_w[256x512] @ S[b][512xHW]
// A tile (weights, shared by all 8 waves) staged via TDM. grid=(18,16,32).
// ---------------------------------------------------------------------------
__global__ __launch_bounds__(256) void proj_kernel(
    const float* __restrict__ sem_in, const float* __restrict__ pw,
    const float* __restrict__ pb, float* __restrict__ sem_out) {
  __shared__ float As[TILE_F];
  const int lane = threadIdx.x & 31;
  const int wave = wave_id();
  const int n0 = (blockIdx.x * 8 + wave) * 16;   // HW tile
  const int m0 = blockIdx.y * 16;                // C1 tile
  const int b  = blockIdx.z;
  const float* Bm = sem_in + (size_t)b * C2N * HWN;   // [C2][HW]
  const int mr   = lane & 15;
  const int nn   = n0 + (lane & 15);
  const int ksel = (lane >> 4) * 2;              // lanes 16-31 hold K+2,K+3
  v8f acc = {};
  for (int kb = 0; kb < C2N; kb += 64) {
    __syncthreads();                             // WAR: tile consumed
    if (wave == 0) {
      tdm_tile_16x64(pw + (size_t)m0 * C2N + kb, 0u, C2N, C1N);
      __builtin_amdgcn_s_wait_tensorcnt(0);
    }
    __syncthreads();                             // tile published
    lds_publish(As);
#pragma unroll
    for (int kk = 0; kk < 64; kk += 4) {
      const int k = kb + kk;
      __builtin_prefetch(&Bm[(size_t)(k + 8) * HWN + nn], 0, 0);
      v2f a, bb;
      a.x  = As[mr * LDS_PITCH + kk + ksel];
      a.y  = As[mr * LDS_PITCH + kk + ksel + 1];
      bb.x = Bm[(size_t)(k + ksel)     * HWN + nn];
      bb.y = Bm[(size_t)(k + ksel + 1) * HWN + nn];
      acc = wmma4(a, bb, acc);
    }
  }
  const int moff = (lane >> 4) * 8;
  float* out = sem_out + (size_t)b * C1N * HWN;
#pragma unroll
  for (int r = 0; r < 8; ++r) {
    const int m = m0 + moff + r;
    out[(size_t)m * HWN + nn] = acc[r] + pb[m];
  }
}

// ---------------------------------------------------------------------------
// Kernel 2: attention logits.  attn[b][c][d] = sum_n q[c,n] * kv[d,n]
// Both operands are K-major row gathers -> TDM-stage A (shared) and per-wave
// B tiles. LDS: [A | B(wave0..7)] = 9 tiles. grid=(2,16,32).
// ---------------------------------------------------------------------------
__global__ __launch_bounds__(256) void attn_kernel(
    const float* __restrict__ vis, const float* __restrict__ sem,
    float* __restrict__ attn) {
  __shared__ float lds[TILE_F * 9];
  const int lane = threadIdx.x & 31;
  const int wave = wave_id();
  const int n0 = (blockIdx.x * 8 + wave) * 16;   // d tile
  const int m0 = blockIdx.y * 16;                // c tile
  const int b  = blockIdx.z;
  const float* Q  = vis + (size_t)b * C1N * HWN;
  const float* KV = sem + (size_t)b * C1N * HWN;
  float* Bs = lds + TILE_F * (1 + wave);
  const unsigned bs_off = (unsigned)(TILE_F * (1 + wave) * sizeof(float));
  const int mr   = lane & 15;
  const int ksel = (lane >> 4) * 2;
  v8f acc = {};
  for (int kb = 0; kb < HWN; kb += 64) {
    __syncthreads();
    if (wave == 0)
      tdm_tile_16x64(Q + (size_t)m0 * HWN + kb, 0u, HWN, C1N);
    tdm_tile_16x64(KV + (size_t)n0 * HWN + kb, bs_off, HWN, C1N);
    __builtin_amdgcn_s_wait_tensorcnt(0);        // own B (and A for wave 0)
    __syncthreads();
    lds_publish(lds);
#pragma unroll
    for (int kk = 0; kk < 64; kk += 4) {
      v2f a, bb;
      a.x  = lds[mr * LDS_PITCH + kk + ksel];
      a.y  = lds[mr * LDS_PITCH + kk + ksel + 1];
      bb.x = Bs[mr * LDS_PITCH + kk + ksel];     // B[k][n] = kv[n][k]
      bb.y = Bs[mr * LDS_PITCH + kk + ksel + 1];
      acc = wmma4(a, bb, acc);
    }
  }
  const int moff = (lane >> 4) * 8;
  float* out = attn + (size_t)b * C1N * C1N;
#pragma unroll
  for (int r = 0; r < 8; ++r)
    out[(size_t)(m0 + moff + r) * C1N + (n0 + mr)] = acc[r];
}

// ---------------------------------------------------------------------------
// Kernel 3: softmax over last dim (rows of length 256). One wave32 per row.
// ---------------------------------------------------------------------------
__global__ __launch_bounds__(256) void softmax_kernel(float* __restrict__ attn) {
  const int lane = threadIdx.x & 31;
  const int wave = threadIdx.x >> 5;
  const int row  = blockIdx.x * 8 + wave;        // 0 .. 8191
  float* p = attn + (size_t)row * C1N;
  float v[8];
  float mx = -INFINITY;
#pragma unroll
  for (int i = 0; i < 8; ++i) { v[i] = p[lane + i * 32]; mx = fmaxf(mx, v[i]); }
#pragma unroll
  for (int off = 16; off > 0; off >>= 1) mx = fmaxf(mx, __shfl_xor(mx, off, 32));
  float s = 0.f;
#pragma unroll
  for (int i = 0; i < 8; ++i) { v[i] = __expf(v[i] - mx); s += v[i]; }
#pragma unroll
  for (int off = 16; off > 0; off >>= 1) s += __shfl_xor(s, off, 32);
  const float inv = 1.0f / s;
#pragma unroll
  for (int i = 0; i < 8; ++i) p[lane + i * 32] = v[i] * inv;
}

// ---------------------------------------------------------------------------
// Kernel 4: info = attn @ kv ; fused = gamma*info + visual.  grid=(18,16,32).
// A = softmaxed attn rows, TDM-staged; B (kv) loads are n-coalesced.
// ---------------------------------------------------------------------------
__global__ __launch_bounds__(256) void fuse_kernel(
    const float* __restrict__ attn, const float* __restrict__ sem,
    const float* __restrict__ vis, const float* __restrict__ gamma,
    float* __restrict__ fused) {
  __shared__ float As[TILE_F];
  const int lane = threadIdx.x & 31;
  const int wave = wave_id();
  const int n0 = (blockIdx.x * 8 + wave) * 16;
  const int m0 = blockIdx.y * 16;
  const int b  = blockIdx.z;
  const float* A  = attn + (size_t)b * C1N * C1N;   // [C][C]
  const float* Bm = sem  + (size_t)b * C1N * HWN;   // [C][HW]
  const int mr   = lane & 15;
  const int nn   = n0 + (lane & 15);
  const int ksel = (lane >> 4) * 2;
  v8f acc = {};
  for (int kb = 0; kb < C1N; kb += 64) {
    __syncthreads();
    if (wave == 0) {
      tdm_tile_16x64(A + (size_t)m0 * C1N + kb, 0u, C1N, C1N);
      __builtin_amdgcn_s_wait_tensorcnt(0);
    }
    __syncthreads();
    lds_publish(As);
#pragma unroll
    for (int kk = 0; kk < 64; kk += 4) {
      const int k = kb + kk;
      __builtin_prefetch(&Bm[(size_t)(k + 8) * HWN + nn], 0, 0);
      v2f a, bb;
      a.x  = As[mr * LDS_PITCH + kk + ksel];
      a.y  = As[mr * LDS_PITCH + kk + ksel + 1];
      bb.x = Bm[(size_t)(k + ksel)     * HWN + nn];
      bb.y = Bm[(size_t)(k + ksel + 1) * HWN + nn];
      acc = wmma4(a, bb, acc);
    }
  }
  const float g = gamma[0];
  const int moff = (lane >> 4) * 8;
  const float* V = vis   + (size_t)b * C1N * HWN;
  float*       F = fused + (size_t)b * C1N * HWN;
#pragma unroll
  for (int r = 0; r < 8; ++r) {
    const int m = m0 + moff + r;
    F[(size_t)m * HWN + nn] = g * acc[r] + V[(size_t)m * HWN + nn];
  }
}

// ---------------------------------------------------------------------------
// Kernel 5: conv3x3 as implicit GEMM. M=co, N=hw, K=ci*9 (=2304), SAME padding
// via per-lane boundary selects (v_cndmask; EXEC stays all-ones for WMMA).
// Weight tile (row-major co x (ci*3*3)) TDM-staged per 64-wide K slab.
// ---------------------------------------------------------------------------
__device__ __forceinline__ float im2col_elem(const float* __restrict__ F,
                                             int k, int y, int x) {
  const int ci = k / 9;
  const int r  = k - ci * 9;
  const int sy = y + r / 3 - 1;
  const int sx = x + (r % 3) - 1;
  const bool ok = ((unsigned)sy < (unsigned)WDIM) && ((unsigned)sx < (unsigned)WDIM);
  return ok ? F[(size_t)ci * HWN + sy * WDIM + sx] : 0.0f;
}

__global__ __launch_bounds__(256) void conv_kernel(
    const float* __restrict__ fused, const float* __restrict__ cw,
    const float* __restrict__ cb, float* __restrict__ ybuf) {
  __shared__ float As[TILE_F];
  const int lane = threadIdx.x & 31;
  const int wave = wave_id();
  const int n0 = (blockIdx.x * 8 + wave) * 16;
  const int m0 = blockIdx.y * 16;
  const int b  = blockIdx.z;
  const float* F = fused + (size_t)b * C1N * HWN;
  const int mr   = lane & 15;
  const int nn   = n0 + (lane & 15);
  const int yy   = nn / WDIM;
  const int xx   = nn - yy * WDIM;
  const int ksel = (lane >> 4) * 2;
  v8f acc = {};
  for (int kb = 0; kb < KCONV; kb += 64) {
    __syncthreads();
    if (wave == 0) {
      tdm_tile_16x64(cw + (size_t)m0 * KCONV + kb, 0u, KCONV, C1N);
      __builtin_amdgcn_s_wait_tensorcnt(0);
    }
    __syncthreads();
    lds_publish(As);
#pragma unroll
    for (int kk = 0; kk < 64; kk += 4) {
      const int k = kb + kk;
      v2f a, bb;
      a.x  = As[mr * LDS_PITCH + kk + ksel];
      a.y  = As[mr * LDS_PITCH + kk + ksel + 1];
      bb.x = im2col_elem(F, k + ksel,     yy, xx);
      bb.y = im2col_elem(F, k + ksel + 1, yy, xx);
      acc = wmma4(a, bb, acc);
    }
  }
  const int moff = (lane >> 4) * 8;
  float* out = ybuf + (size_t)b * C1N * HWN;
#pragma unroll
  for (int r = 0; r < 8; ++r) {
    const int m = m0 + moff + r;
    out[(size_t)m * HWN + nn] = acc[r] + cb[m];
  }
}

// ---------------------------------------------------------------------------
// Kernel 6: BN batch stats per channel (mean, rstd) over B*H*W = 73728 elems.
// float4 (b128) loads — this pass sits directly on the HBM roofline.
// ---------------------------------------------------------------------------
__global__ __launch_bounds__(256) void bnstats_kernel(
    const float* __restrict__ ybuf, float* __restrict__ mean,
    float* __restrict__ rstd) {
  const int c = blockIdx.x;
  const int t = threadIdx.x;
  float s1 = 0.f, s2 = 0.f;
  for (int j = t; j < NB * HWN / 4; j += 256) {
    const int idx4 = j * 4;
    const int b  = idx4 / HWN;
    const int hw = idx4 - b * HWN;
    const float4 v = *(const float4*)&ybuf[((size_t)b * C1N + c) * HWN + hw];
    s1 += v.x + v.y + v.z + v.w;
    s2 += v.x * v.x + v.y * v.y + v.z * v.z + v.w * v.w;
  }
  __shared__ float sh1[256], sh2[256];
  sh1[t] = s1; sh2[t] = s2;
  __syncthreads();
  for (int off = 128; off > 0; off >>= 1) {
    if (t < off) { sh1[t] += sh1[t + off]; sh2[t] += sh2[t + off]; }
    __syncthreads();
  }
  if (t == 0) {
    const float invN = 1.0f / (float)(NB * HWN);
    const float m = sh1[0] * invN;
    const float var = sh2[0] * invN - m * m;
    mean[c] = m;
    rstd[c] = rsqrtf(var + 1e-5f);
  }
}

// ---------------------------------------------------------------------------
// Kernel 7: y_hat = relu((y - mean)*rstd*scale + bias), float4 per thread.
// A float4 never crosses a (b,c) row since HW = 2304 is a multiple of 4.
// ---------------------------------------------------------------------------
__global__ __launch_bounds__(256) void bnapply_kernel(
    const float* __restrict__ ybuf, const float* __restrict__ mean,
    const float* __restrict__ rstd, const float* __restrict__ scale,
    const float* __restrict__ bias, float* __restrict__ out) {
  const size_t i4 = ((size_t)blockIdx.x * 256 + threadIdx.x) * 4;
  const int c = (int)((i4 / HWN) % C1N);
  const float a = rstd[c] * scale[c];
  const float b = bias[c] - mean[c] * a;
  float4 v = *(const float4*)&ybuf[i4];
  v.x = fmaxf(v.x * a + b, 0.f);
  v.y = fmaxf(v.y * a + b, 0.f);
  v.z = fmaxf(v.z * a + b, 0.f);
  v.w = fmaxf(v.w * a + b, 0.f);
  *(float4*)&out[i4] = v;
}

// ---------------------------------------------------------------------------
extern "C" void kernel_launch(void* const* d_in, const int* in_sizes, int n_in,
                              void* d_out, int out_size, void* d_ws, size_t ws_size,
                              hipStream_t stream) {
  (void)in_sizes; (void)n_in; (void)out_size; (void)ws_size;
  const float* vis   = (const float*)d_in[0];
  const float* semin = (const float*)d_in[1];
  const float* pw    = (const float*)d_in[2];
  const float* pb    = (const float*)d_in[3];
  const float* gamma = (const float*)d_in[4];
  const float* cw    = (const float*)d_in[5];
  const float* cb    = (const float*)d_in[6];
  const float* bns   = (const float*)d_in[7];
  const float* bnb   = (const float*)d_in[8];
  float* out = (float*)d_out;

  // Workspace layout (floats). sem buffer is reused for pre-BN y after kernel 4.
  float* ws    = (float*)d_ws;
  float* sem   = ws;                              // 32*256*2304 = 18,874,368
  float* attn  = sem  + (size_t)NB * C1N * HWN;   // 32*256*256  =  2,097,152
  float* fused = attn + (size_t)NB * C1N * C1N;   // 18,874,368
  float* mean  = fused + (size_t)NB * C1N * HWN;  // 256
  float* rstd  = mean + C1N;                      // 256
  float* ybuf  = sem;                             // reuse (sem dead after fuse)

  const dim3 blk(256);
  proj_kernel   <<<dim3(18, 16, NB), blk, 0, stream>>>(semin, pw, pb, sem);
  attn_kernel   <<<dim3(2, 16, NB),  blk, 0, stream>>>(vis, sem, attn);
  softmax_kernel<<<dim3(1024),       blk, 0, stream>>>(attn);
  fuse_kernel   <<<dim3(18, 16, NB), blk, 0, stream>>>(attn, sem, vis, gamma, fused);
  conv_kernel   <<<dim3(18, 16, NB), blk, 0, stream>>>(fused, cw, cb, ybuf);
  bnstats_kernel<<<dim3(C1N),        blk, 0, stream>>>(ybuf, mean, rstd);
  bnapply_kernel<<<dim3(NB * C1N * HWN / 1024), blk, 0, stream>>>(
      ybuf, mean, rstd, bns, bnb, out);
}